// Neuron_80212809220201
// MI455X (gfx1250) — compile-verified
//
#include <hip/hip_runtime.h>

typedef __attribute__((ext_vector_type(2))) float v2f;
typedef __attribute__((ext_vector_type(8))) float v8f;

#define NN      16384
#define WAVES   8
#define KCHUNK  (NN / WAVES)   /* 2048 columns per wave */

// One workgroup = 16 output rows. 8 waves split the K dimension; each wave
// accumulates a 16x16 f32 tile with V_WMMA_F32_16X16X4_F32 where B broadcasts
// the state chunk into all 16 columns, so every column of D holds the same
// 16 row dot-products. Partial sums are reduced across waves through LDS.
__global__ __launch_bounds__(256) void hopfield_gemv_wmma(
    const float* __restrict__ W, const float* __restrict__ s,
    const float* __restrict__ bias, float* __restrict__ out) {
  __shared__ float red[WAVES * 16];

  const int tid  = threadIdx.x;
  const int wave = tid >> 5;
  const int lane = tid & 31;
  const int lm   = lane & 15;   // row within tile (A layout: lanes 0-15 == lanes 16-31 rows)
  const int h    = lane >> 4;   // half-wave: selects K=0,1 vs K=2,3 of the 16x4 A tile

  const int rowBase = blockIdx.x * 16;
  const int kStart  = wave * KCHUNK;

  // A-layout (ISA 7.12.2, 32-bit A 16x4): lane m v0=K0,v1=K1; lane m+16 v0=K2,v1=K3.
  const float* wp = W + (size_t)(rowBase + lm) * NN + kStart + 2 * h;
  // B broadcast: value depends only on K; mirror A's half-wave K split.
  const float* sp = s + kStart + 2 * h;

  v8f acc = {};
#pragma unroll 4
  for (int k = 0; k < KCHUNK; k += 4) {
    v2f a = *(const v2f*)(wp + k);   // W[row][k+2h], W[row][k+2h+1]
    v2f b = *(const v2f*)(sp + k);   // s[k+2h],      s[k+2h+1]
    // 8 args: (neg_a, A, neg_b, B, c_mod, C, reuse_a, reuse_b)
    acc = __builtin_amdgcn_wmma_f32_16x16x4_f32(
        false, a, false, b, (short)0, acc, false, false);
  }

  // D layout: VGPR v -> row rowBase+v (lane 0) / rowBase+8+v (lane 16);
  // all 16 columns identical, so lanes 0 and 16 carry the full result.
  if (lm == 0) {
    float* dst = &red[wave * 16 + h * 8];
#pragma unroll
    for (int v = 0; v < 8; ++v) dst[v] = acc[v];
  }
  __syncthreads();

  if (tid < 16) {
    const int row = rowBase + tid;
    float sum = 0.0f;
#pragma unroll
    for (int w = 0; w < WAVES; ++w) sum += red[w * 16 + tid];
    const float sv   = s[row];
    const float diag = W[(size_t)row * NN + row];
    const float act  = sum - diag * sv + (float)(NN - 1) * bias[row];
    out[row]      = act;                              // activation
    out[NN + row] = (act >= 0.0f) ? 1.0f : -1.0f;     // new state
  }
}

extern "C" void kernel_launch(void* const* d_in, const int* in_sizes, int n_in,
                              void* d_out, int out_size, void* d_ws, size_t ws_size,
                              hipStream_t stream) {
  const float* W    = (const float*)d_in[0];   // (N, N) f32
  const float* s    = (const float*)d_in[1];   // (N,)  f32
  const float* bias = (const float*)d_in[2];   // (N,)  f32
  float* out        = (float*)d_out;           // act (N) ++ new_state (N)

  dim3 grid(NN / 16);   // 1024 row tiles
  dim3 block(256);      // 8 wave32 per tile, splitting K
  hopfield_gemv_wmma<<<grid, block, 0, stream>>>(W, s, bias, out);
}